// GIPAWideConv_37074157699188
// MI455X (gfx1250) — compile-verified
//
#include <hip/hip_runtime.h>
#include <hip/hip_bf16.h>

typedef __attribute__((ext_vector_type(2))) float v2f;
typedef __attribute__((ext_vector_type(8))) float v8f;

#define NCOLS 128          // NUM_HEADS * OUT_FEATS
#define NODE_K 256         // IN_FEATS
#define EDGE_K 64          // EDGE_FEATS
#define NHEADS 4

// ---------------------------------------------------------------------------
// Generic row-major f32 GEMM: C[M,128] = A[M,K] @ B[K,128]
// Block = 256 threads = 8 waves; block computes a 16-row stripe, each wave a
// 16x16 tile via V_WMMA_F32_16X16X4_F32. A panel staged in LDS (shared by all
// 8 waves); B (the small weight matrix) streams through L2.
// ---------------------------------------------------------------------------
template <int K>
__global__ void __launch_bounds__(256)
gemm_f32_wmma(const float* __restrict__ A, const float* __restrict__ B,
              float* __restrict__ C, int M) {
    __shared__ float Apanel[16 * K];
    const int lane = threadIdx.x & 31;
    const int wave = threadIdx.x >> 5;
    const int m0   = blockIdx.x * 16;

    // cooperative stage of the 16xK A panel (clamped for safety at the tail)
    for (int idx = threadIdx.x; idx < 16 * K; idx += 256) {
        int r = idx / K, c = idx % K;
        int row = m0 + r; row = row < M ? row : M - 1;
        Apanel[idx] = A[(size_t)row * K + c];
    }
    __syncthreads();

    const int n0      = wave * 16;        // this wave's column tile
    const int half    = lane >> 4;        // 0: K pair {0,1}, 1: K pair {2,3}
    const int l15     = lane & 15;
    const int kpair   = half * 2;

    v8f acc = {};
    for (int k = 0; k < K; k += 4) {
        v2f a, b;
        a.x = Apanel[l15 * K + k + kpair];
        a.y = Apanel[l15 * K + k + kpair + 1];
        b.x = B[(size_t)(k + kpair)     * NCOLS + n0 + l15];
        b.y = B[(size_t)(k + kpair + 1) * NCOLS + n0 + l15];
        acc = __builtin_amdgcn_wmma_f32_16x16x4_f32(
            false, a, false, b, (short)0, acc, false, false);
    }

    // C/D layout: VGPR r -> row (r + 8*half), column n0 + (lane&15)
    const int mbase = half * 8;
#pragma unroll
    for (int r = 0; r < 8; ++r) {
        int row = m0 + mbase + r;
        if (row < M) C[(size_t)row * NCOLS + n0 + l15] = acc[r];
    }
}

// ---------------------------------------------------------------------------
// Fused edge projection + attention score.
// Per block: 16 edges. Each of 8 waves computes the 16x16 tile of
// e = edge_feat @ W_edge for its column range with WMMA, then (without ever
// storing e) multiplies in-register by gathered h[src] values, reduces across
// the 16-lane halves, and the block combines the two column-tiles per head:
//   score[e][h] = leaky_relu( dot(e[e,h,:], h[src[e],h,:]) / sqrt(32) )
// ---------------------------------------------------------------------------
__global__ void __launch_bounds__(256)
edge_proj_score(const float* __restrict__ EF, const float* __restrict__ We,
                const float* __restrict__ H, const int* __restrict__ src,
                float* __restrict__ score, int E) {
    __shared__ float Apanel[16 * EDGE_K];   // 4 KB edge-feature panel
    __shared__ int   srcIdx[16];
    __shared__ float spart[8][16];          // per-wave partial scores

    const int lane = threadIdx.x & 31;
    const int wave = threadIdx.x >> 5;
    const int m0   = blockIdx.x * 16;

    for (int idx = threadIdx.x; idx < 16 * EDGE_K; idx += 256) {
        int r = idx >> 6, c = idx & 63;
        int row = m0 + r; row = row < E ? row : E - 1;
        Apanel[idx] = EF[(size_t)row * EDGE_K + c];
    }
    if (threadIdx.x < 16) {
        int e = m0 + threadIdx.x;
        srcIdx[threadIdx.x] = src[e < E ? e : E - 1];
    }
    __syncthreads();

    const int n0    = wave * 16;
    const int half  = lane >> 4;
    const int l15   = lane & 15;
    const int kpair = half * 2;

    v8f acc = {};
    for (int k = 0; k < EDGE_K; k += 4) {
        v2f a, b;
        a.x = Apanel[l15 * EDGE_K + k + kpair];
        a.y = Apanel[l15 * EDGE_K + k + kpair + 1];
        b.x = We[(size_t)(k + kpair)     * NCOLS + n0 + l15];
        b.y = We[(size_t)(k + kpair + 1) * NCOLS + n0 + l15];
        acc = __builtin_amdgcn_wmma_f32_16x16x4_f32(
            false, a, false, b, (short)0, acc, false, false);
    }

    // Fused dot with gathered h[src]: lane holds e[row][n0+l15] in acc[r]
    // where row = mbase + r. Gather matching h element (L2-resident) and
    // reduce the 16 columns of this tile across the 16-lane half.
    const int mbase = half * 8;
    float prod[8];
#pragma unroll
    for (int r = 0; r < 8; ++r) {
        int s = srcIdx[mbase + r];
        float hv = H[(size_t)s * NCOLS + n0 + l15];
        prod[r] = acc[r] * hv;
    }
#pragma unroll
    for (int off = 1; off < 16; off <<= 1) {
#pragma unroll
        for (int r = 0; r < 8; ++r) prod[r] += __shfl_xor(prod[r], off, 32);
    }
    if (l15 == 0) {
#pragma unroll
        for (int r = 0; r < 8; ++r) spart[wave][mbase + r] = prod[r];
    }
    __syncthreads();

    // head h spans column tiles of waves 2h and 2h+1
    if (threadIdx.x < 64) {
        int m = threadIdx.x >> 2, hd = threadIdx.x & 3;
        int e = m0 + m;
        if (e < E) {
            float sv = (spart[2 * hd][m] + spart[2 * hd + 1][m]) * 0.17677669529663689f;
            sv = sv > 0.f ? sv : 0.01f * sv;   // leaky_relu, slope 0.01
            score[(size_t)e * NHEADS + hd] = sv;
        }
    }
}

// ---------------------------------------------------------------------------
// Softmax machinery
// ---------------------------------------------------------------------------
__global__ void __launch_bounds__(256)
init_kernel(float* __restrict__ out, float* __restrict__ smax,
            float* __restrict__ denom, int nOut, int nSeg) {
    int gid = blockIdx.x * 256 + threadIdx.x;
    if (gid < nOut) out[gid] = 0.f;
    if (gid < nSeg) { smax[gid] = __int_as_float(0xFF800000); denom[gid] = 0.f; }
}

__device__ __forceinline__ void atomicMaxFloat(float* addr, float val) {
    // order-preserving int punning: works for mixed signs with -inf init
    if (val >= 0.f) atomicMax((int*)addr, __float_as_int(val));
    else            atomicMin((unsigned int*)addr, __float_as_uint(val));
}

__global__ void __launch_bounds__(256)
segmax_kernel(const float* __restrict__ score, const int* __restrict__ dst,
              float* __restrict__ smax, int E) {
    int gid = blockIdx.x * 256 + threadIdx.x;
    if (gid >= E * NHEADS) return;
    int e = gid >> 2, hd = gid & 3;
    atomicMaxFloat(&smax[dst[e] * NHEADS + hd], score[gid]);
}

__global__ void __launch_bounds__(256)
expsum_kernel(const float* __restrict__ score, const int* __restrict__ dst,
              const float* __restrict__ smax, float* __restrict__ aarr,
              float* __restrict__ denom, int E) {
    int gid = blockIdx.x * 256 + threadIdx.x;
    if (gid >= E * NHEADS) return;
    int e = gid >> 2, hd = gid & 3;
    float v = __expf(score[gid] - smax[dst[e] * NHEADS + hd]);
    aarr[gid] = v;
    atomicAdd(&denom[dst[e] * NHEADS + hd], v);
}

// ---------------------------------------------------------------------------
// Weighted aggregation: out[dst] += h[src] * attn  (one thread per edge-feat)
// ---------------------------------------------------------------------------
__global__ void __launch_bounds__(256)
aggregate_kernel(const float* __restrict__ H, const int* __restrict__ src,
                 const int* __restrict__ dst, const float* __restrict__ aarr,
                 const float* __restrict__ denom, float* __restrict__ out, int E) {
    int gid = blockIdx.x * 256 + threadIdx.x;
    int e = gid >> 7;            // / 128 feats
    if (e >= E) return;
    int f  = gid & 127;
    int hd = f >> 5;
    int s = src[e], d = dst[e];
    float attn = aarr[e * NHEADS + hd] / denom[d * NHEADS + hd];
    float val  = H[(size_t)s * NCOLS + f] * attn;
    atomicAdd(&out[(size_t)d * NCOLS + f], val);
}

// ---------------------------------------------------------------------------
extern "C" void kernel_launch(void* const* d_in, const int* in_sizes, int n_in,
                              void* d_out, int out_size, void* d_ws, size_t ws_size,
                              hipStream_t stream) {
    const float* node_feat = (const float*)d_in[0];
    const float* edge_feat = (const float*)d_in[1];
    const int*   src       = (const int*)d_in[2];
    const int*   dst       = (const int*)d_in[3];
    const float* W_node    = (const float*)d_in[4];
    const float* W_edge    = (const float*)d_in[5];
    float*       out       = (float*)d_out;

    const int N = in_sizes[0] / NODE_K;   // 50000
    const int E = in_sizes[2];            // 800000

    // workspace carve-out (floats): h | score | a | smax | denom
    float* h     = (float*)d_ws;
    float* score = h     + (size_t)N * NCOLS;
    float* aarr  = score + (size_t)E * NHEADS;
    float* smax  = aarr  + (size_t)E * NHEADS;
    float* denom = smax  + (size_t)N * NHEADS;

    // 1) h = node_feat @ W_node  (WMMA f32)
    gemm_f32_wmma<NODE_K><<<(N + 15) / 16, 256, 0, stream>>>(node_feat, W_node, h, N);

    // 2) fused e-projection + score (WMMA f32, e never materialized)
    edge_proj_score<<<(E + 15) / 16, 256, 0, stream>>>(edge_feat, W_edge, h, src, score, E);

    // 3) init out/smax/denom
    int nOut = N * NCOLS;
    init_kernel<<<(nOut + 255) / 256, 256, 0, stream>>>(out, smax, denom, nOut, N * NHEADS);

    // 4) segment max
    segmax_kernel<<<(E * NHEADS + 255) / 256, 256, 0, stream>>>(score, dst, smax, E);

    // 5) exp + segment sum
    expsum_kernel<<<(E * NHEADS + 255) / 256, 256, 0, stream>>>(score, dst, smax, aarr, denom, E);

    // 6) weighted aggregation into out
    long long total = (long long)E * NCOLS;
    aggregate_kernel<<<(int)((total + 255) / 256), 256, 0, stream>>>(h, src, dst, aarr, denom, out, E);
}